// GVPPairEnergies_43430709297207
// MI455X (gfx1250) — compile-verified
//
#include <hip/hip_runtime.h>
#include <stdint.h>

#define EPSN 1e-8f
#define HASH_SIZE (1 << 19)   // 524288 slots > 2 * 152000 edges

typedef __attribute__((ext_vector_type(2))) float v2f;
typedef __attribute__((ext_vector_type(8))) float v8f;

// ---------------------------------------------------------------------------
// A-operand: concat of up to 4 row-major segments, each optionally gathered
// through an int64 index.  Each segment's K-extent is padded to a multiple of
// 4 and the weight matrix is pre-packed with matching zero rows, so the GEMM
// inner loop needs NO per-element guards and NO divergent branches.
// ---------------------------------------------------------------------------
struct Seg   { const float* base; const long long* gather; int stride; int wpad; };
struct ASpec { Seg seg[4]; };

// ---------------------------------------------------------------------------
// Pack W[K,N] (row major, segment widths w0..w3 along K) into fragment order:
//   pair-row (g*2 + half) of Npad float2's; element v of pair = padded row
//   kp = 4*g + 2*half + v.  Zero rows inserted at segment pads, zero columns
//   beyond N (Npad = ceil32(N) for NT=2 tiles).
// ---------------------------------------------------------------------------
__global__ void k_pack_w(const float* __restrict__ W, int N, int Npad,
                         int w0, int w1, int w2, int w3,
                         float* __restrict__ Wp2, int Kp) {
    long long t = (long long)blockIdx.x * blockDim.x + threadIdx.x;
    if (t >= (long long)Kp * Npad) return;
    int kp = (int)(t / Npad);
    int n  = (int)(t % Npad);
    int wd[4] = {w0, w1, w2, w3};
    int rem = kp, korig = 0;
    bool valid = false;
#pragma unroll
    for (int s = 0; s < 4; ++s) {
        int wp = (wd[s] + 3) & ~3;
        if (rem < wp) { if (rem < wd[s]) { valid = true; korig += rem; } break; }
        rem -= wp; korig += wd[s];
    }
    float val = (valid && n < N) ? W[(long long)korig * N + n] : 0.f;
    long long dst = ((long long)((kp >> 2) * 2 + ((kp >> 1) & 1)) * Npad + n) * 2 + (kp & 1);
    Wp2[dst] = val;
}

// ---------------------------------------------------------------------------
// fp32 WMMA GEMM: C[M,N] = act(A[M,K] * W[K,N] + bias)
// wave = one 16-row M tile x NT 16-col N tiles, V_WMMA_F32_16X16X4_F32, K step 4.
// Fragment layout (ISA 7.12.2): A/B element v, lane half h -> K = k0 + 2h + v;
// C vgpr r -> M = r + 8h, N = lane&15.
// ---------------------------------------------------------------------------
template <int NSEG, int NT>
__global__ void k_gemm_wmma(ASpec a, const float* __restrict__ Wp2,
                            const float* __restrict__ bias,
                            float* __restrict__ C,
                            long long M, int N, int Npad, int ldc, int relu) {
    const int lane = threadIdx.x & 31;
    const int wave = threadIdx.x >> 5;
    const long long mtile = (long long)blockIdx.y * (blockDim.x >> 5) + wave;
    const long long m0 = mtile * 16;
    const int n0 = blockIdx.x * (16 * NT);
    if (m0 >= M) return;                      // wave-uniform

    const int half = lane >> 4;
    const int l15  = lane & 15;
    long long rowA = m0 + l15;
    if (rowA >= M) rowA = M - 1;              // clamp; store is guarded

    // hoist gather + row base per segment (one index load each, outside K loop)
    const float* ap[NSEG];
#pragma unroll
    for (int s = 0; s < NSEG; ++s) {
        long long r = a.seg[s].gather ? a.seg[s].gather[rowA] : rowA;
        ap[s] = a.seg[s].base + r * (long long)a.seg[s].stride + 2 * half;
    }

    const v2f* bpairs = (const v2f*)Wp2;
    v8f acc[NT];
#pragma unroll
    for (int t = 0; t < NT; ++t) acc[t] = (v8f){0.f,0.f,0.f,0.f,0.f,0.f,0.f,0.f};

    int kbase = 0;
#pragma unroll
    for (int s = 0; s < NSEG; ++s) {
        const float* __restrict__ as = ap[s];
        const int wp = a.seg[s].wpad;
        for (int kk = 0; kk < wp; kk += 4) {
            v2f af;
            af.x = as[kk];
            af.y = as[kk + 1];
            const long long g2 = (long long)(((kbase + kk) >> 2) * 2 + half);
            __builtin_prefetch(&bpairs[(g2 + 2) * Npad + n0 + l15], 0, 3);
#pragma unroll
            for (int t = 0; t < NT; ++t) {
                v2f bf = bpairs[g2 * Npad + n0 + t * 16 + l15];
                acc[t] = __builtin_amdgcn_wmma_f32_16x16x4_f32(
                             false, af, false, bf, (short)0, acc[t], false, false);
            }
        }
        kbase += wp;
    }

#pragma unroll
    for (int t = 0; t < NT; ++t) {
#pragma unroll
        for (int r = 0; r < 8; ++r) {
            long long m = m0 + half * 8 + r;
            int n = n0 + t * 16 + l15;
            if (m < M && n < N) {
                float val = acc[t][r] + bias[n];
                if (relu) val = fmaxf(val, 0.f);
                C[m * (long long)ldc + n] = val;
            }
        }
    }
}

// ---------------------------------------------------------------------------
// Vector-channel matmul (VALU; K<=33, too skinny for WMMA):
//   out[row][o][c] = sum_i in(row,i,c) * Wm[i][o]   (in = concat of <=3 segs)
// act!=0: vector_act=relu => multiply by own norm (norm >= sqrt(EPS) > 0).
// ---------------------------------------------------------------------------
struct VSeg { const float* base; const long long* gather; int width; };

__global__ void k_vec_mm(VSeg g0, VSeg g1, VSeg g2, int VI,
                         const float* __restrict__ Wm, int VO,
                         float* __restrict__ out, long long M, int act) {
    long long t = (long long)blockIdx.x * blockDim.x + threadIdx.x;
    if (t >= M * (long long)VO) return;
    long long row = t / VO;
    int o = (int)(t % VO);
    VSeg segs[3] = {g0, g1, g2};
    float ax = 0.f, ay = 0.f, az = 0.f;
    int ig = 0;
    for (int s = 0; s < 3; ++s) {
        if (segs[s].width == 0) continue;
        long long r = segs[s].gather ? segs[s].gather[row] : row;
        const float* p = segs[s].base + r * (long long)segs[s].width * 3;
        for (int i = 0; i < segs[s].width; ++i, ++ig) {
            float w = Wm[ig * VO + o];
            ax = fmaf(p[i * 3 + 0], w, ax);
            ay = fmaf(p[i * 3 + 1], w, ay);
            az = fmaf(p[i * 3 + 2], w, az);
        }
    }
    if (act) {
        float nrm = sqrtf(fmaxf(ax * ax + ay * ay + az * az, EPSN));
        ax *= nrm; ay *= nrm; az *= nrm;
    }
    float* q = out + (row * VO + o) * 3;
    q[0] = ax; q[1] = ay; q[2] = az;
}

__global__ void k_norm3(const float* __restrict__ vh, float* __restrict__ vn, long long cnt) {
    long long t = (long long)blockIdx.x * blockDim.x + threadIdx.x;
    if (t >= cnt) return;
    float x = vh[t * 3], y = vh[t * 3 + 1], z = vh[t * 3 + 2];
    vn[t] = sqrtf(fmaxf(x * x + y * y + z * z, EPSN));
}

// scalar layernorm, wave-per-row, wave32 butterfly reduction
__global__ void k_ln_s(float* __restrict__ s, const float* __restrict__ gamma,
                       const float* __restrict__ beta, long long M, int S) {
    long long row = (long long)blockIdx.x * (blockDim.x >> 5) + (threadIdx.x >> 5);
    int lane = threadIdx.x & 31;
    if (row >= M) return;
    float* p = s + row * (long long)S;
    float sum = 0.f, sq = 0.f;
    for (int c = lane; c < S; c += 32) { float x = p[c]; sum += x; sq += x * x; }
#pragma unroll
    for (int m = 16; m > 0; m >>= 1) {
        sum += __shfl_xor(sum, m, 32);
        sq  += __shfl_xor(sq,  m, 32);
    }
    float mu  = sum / (float)S;
    float var = sq / (float)S - mu * mu;
    float inv = rsqrtf(var + 1e-5f);
    for (int c = lane; c < S; c += 32)
        p[c] = (p[c] - mu) * inv * gamma[c] + beta[c];
}

// vector layernorm: v /= sqrt(mean_i(max(|v_i|^2, EPS)))
__global__ void k_ln_v(float* __restrict__ v, long long M, int NV) {
    long long row = (long long)blockIdx.x * blockDim.x + threadIdx.x;
    if (row >= M) return;
    float* p = v + row * (long long)NV * 3;
    float acc = 0.f;
    for (int i = 0; i < NV; ++i) {
        float x = p[i * 3], y = p[i * 3 + 1], z = p[i * 3 + 2];
        acc += fmaxf(x * x + y * y + z * z, EPSN);
    }
    float inv = rsqrtf(acc / (float)NV);
    for (int j = 0; j < NV * 3; ++j) p[j] *= inv;
}

__global__ void k_fill(float* __restrict__ p, long long n, float v) {
    long long t = (long long)blockIdx.x * blockDim.x + threadIdx.x;
    if (t < n) p[t] = v;
}
__global__ void k_add(float* __restrict__ d, const float* __restrict__ s, long long n) {
    long long t = (long long)blockIdx.x * blockDim.x + threadIdx.x;
    if (t < n) d[t] += s[t];
}
__global__ void k_deg(const long long* __restrict__ dst, float* __restrict__ deg, long long E) {
    long long e = (long long)blockIdx.x * blockDim.x + threadIdx.x;
    if (e < E) atomicAdd(&deg[dst[e]], 1.f);
}
__global__ void k_scatter(const float* __restrict__ src_, const long long* __restrict__ idx,
                          float* __restrict__ agg, long long E, int C) {
    long long t = (long long)blockIdx.x * blockDim.x + threadIdx.x;
    if (t >= E * (long long)C) return;
    long long e = t / C; int c = (int)(t % C);
    atomicAdd(&agg[idx[e] * (long long)C + c], src_[t]);
}
__global__ void k_resid(float* __restrict__ x, const float* __restrict__ agg,
                        const float* __restrict__ deg, long long M, int C) {
    long long t = (long long)blockIdx.x * blockDim.x + threadIdx.x;
    if (t >= M * (long long)C) return;
    long long m = t / C;
    x[t] += agg[t] / fmaxf(deg[m], 1.f);
}
__global__ void k_merge_rev(float* __restrict__ dst, const float* __restrict__ tmp,
                            const int* __restrict__ rev, long long E, int C) {
    long long t = (long long)blockIdx.x * blockDim.x + threadIdx.x;
    if (t >= E * (long long)C) return;
    long long e = t / C; int c = (int)(t % C);
    dst[t] = 0.5f * (tmp[t] + tmp[(long long)rev[e] * C + c]);
}

// ------------------------- reverse-edge index via hash ----------------------
__global__ void k_hash_init(int* keys, int n) {
    int i = blockIdx.x * blockDim.x + threadIdx.x;
    if (i < n) keys[i] = -1;
}
__global__ void k_hash_insert(const long long* src, const long long* dst, long long E,
                              long long nn, int* keys, int* vals) {
    long long e = (long long)blockIdx.x * blockDim.x + threadIdx.x;
    if (e >= E) return;
    int key = (int)(src[e] * nn + dst[e]);          // < 64e6, fits int, unique
    unsigned h = ((unsigned)key * 2654435761u) & (HASH_SIZE - 1);
    for (;;) {
        int old = atomicCAS(&keys[h], -1, key);
        if (old == -1) { vals[h] = (int)e; break; }
        h = (h + 1) & (HASH_SIZE - 1);
    }
}
__global__ void k_hash_lookup(const long long* src, const long long* dst, long long E,
                              long long nn, const int* keys, const int* vals,
                              int* rev) {
    long long e = (long long)blockIdx.x * blockDim.x + threadIdx.x;
    if (e >= E) return;
    int key = (int)(dst[e] * nn + src[e]);
    unsigned h = ((unsigned)key * 2654435761u) & (HASH_SIZE - 1);
    while (keys[h] != key) h = (h + 1) & (HASH_SIZE - 1);
    rev[e] = vals[h];
}

// Pairwise in-place symmetrization of the final [E,20,20] table + self mask.
__global__ void k_merge_out(float* __restrict__ out, const int* __restrict__ rev,
                            const long long* __restrict__ src,
                            const long long* __restrict__ dst, long long E) {
    long long t = (long long)blockIdx.x * blockDim.x + threadIdx.x;
    if (t >= E * 400LL) return;
    long long e = t / 400;
    int idx = (int)(t % 400);
    int a = idx / 20, b = idx % 20;
    long long r = rev[e];
    if (src[e] == dst[e]) {
        if (a != b) out[t] = 0.f;                    // etab * eye; diag == x_aa
    } else if (e < r) {
        float x1 = out[e * 400 + a * 20 + b];
        float x2 = out[r * 400 + b * 20 + a];
        float m = 0.5f * (x1 + x2);
        out[e * 400 + a * 20 + b] = m;
        out[r * 400 + b * 20 + a] = m;
    }
}

// ============================ host orchestration ============================
struct Pgvp { const float *wh, *ws_b, *ws_w, *wv; };
struct Pln  { const float *beta, *gamma; };
struct GSeg { const float* base; const long long* gather; int width; int stride; };

extern "C" void kernel_launch(void* const* d_in, const int* in_sizes, int n_in,
                              void* d_out, int out_size, void* d_ws, size_t ws_size,
                              hipStream_t stream) {
    const float* hVs = (const float*)d_in[0];
    const float* hVv = (const float*)d_in[1];
    const float* hEs = (const float*)d_in[2];
    const float* hEv = (const float*)d_in[3];
    const long long NN = in_sizes[0] / 6;
    const long long E  = in_sizes[2] / 32;
    const long long* ei  = (const long long*)d_in[n_in - 1];
    const long long* src = ei;
    const long long* dst = ei + E;

    // ---- parameter leaves, JAX tree order (dict keys sorted alphabetically)
    const bool concat = (n_in == 6);
    const float* pbase = (const float*)d_in[4];
    size_t cur = 0; int leaf = 0;
    auto take = [&](size_t n) -> const float* {
        if (concat) { const float* p = pbase + cur; cur += n; return p; }
        return (const float*)d_in[4 + (leaf++)];
    };
    auto gvpP = [&](int si, int vi, int so, int vo) {
        Pgvp p; int h = vi > vo ? vi : vo;
        p.wh = take((size_t)vi * h); p.ws_b = take(so);
        p.ws_w = take((size_t)(si + h) * so);
        p.wv = vo > 0 ? take((size_t)h * vo) : nullptr;
        return p;
    };
    auto lnp = [&](int so) { Pln l; l.beta = take(so); l.gamma = take(so); return l; };

    // top-level sorted: W_e, W_out, W_v, edge_layers, ln_e, ln_v, node_layers
    Pgvp W_e  = gvpP(32, 1, 32, 1);
    Pgvp W_out = gvpP(32, 1, 400, 0);
    Pgvp W_v  = gvpP(6, 3, 100, 16);
    struct EL { Pgvp ff0, ff1; Pln ln0, ln1; Pgvp m0, m1, m2; } el[2];
    for (int l = 0; l < 2; ++l) {   // per-layer sorted: ff, ln0, ln1, msg
        el[l].ff0 = gvpP(32, 1, 128, 2);  el[l].ff1 = gvpP(128, 2, 32, 1);
        el[l].ln0 = lnp(32);              el[l].ln1 = lnp(32);
        el[l].m0 = gvpP(232, 33, 32, 1);  el[l].m1 = gvpP(32, 1, 32, 1);
        el[l].m2 = gvpP(32, 1, 32, 1);
    }
    Pln ln_e = lnp(32);
    Pln ln_v = lnp(100);
    struct NL { Pgvp ff0, ff1; Pln ln0, ln1; Pgvp m0, m1, m2; } nl[2];
    for (int l = 0; l < 2; ++l) {
        nl[l].ff0 = gvpP(100, 16, 400, 32); nl[l].ff1 = gvpP(400, 32, 100, 16);
        nl[l].ln0 = lnp(100);               nl[l].ln1 = lnp(100);
        nl[l].m0 = gvpP(232, 33, 100, 16);  nl[l].m1 = gvpP(100, 16, 100, 16);
        nl[l].m2 = gvpP(100, 16, 100, 16);
    }

    // ---- workspace carve
    char* w = (char*)d_ws;
    auto alloc_f = [&](size_t n) { float* p = (float*)w; w += ((n * 4 + 255) / 256) * 256; return p; };
    auto alloc_i = [&](size_t n) { int*   p = (int*)w;   w += ((n * 4 + 255) / 256) * 256; return p; };
    float* s_node = alloc_f(NN * 100);
    float* v_node = alloc_f(NN * 48);
    float* es     = alloc_f(E * 32);
    float* ev     = alloc_f(E * 3);
    float* agg_s  = alloc_f(NN * 100);
    float* agg_v  = alloc_f(NN * 48);
    float* deg    = alloc_f(NN);
    float* nvh    = alloc_f(NN * 96);
    float* nvn    = alloc_f(NN * 32);
    float* nfs    = alloc_f(NN * 400);
    float* nfv    = alloc_f(NN * 96);
    float* nfs2   = alloc_f(NN * 100);
    float* nfv2   = alloc_f(NN * 48);
    float* hVsp   = alloc_f(NN * 6);   // padded-safe copy of h_V_s
    float* vh     = alloc_f(E * 99);   // up to [E,33,3]
    float* vn     = alloc_f(E * 33);
    float* sA     = alloc_f(E * 128);
    float* sB     = alloc_f(E * 128);
    float* vA     = alloc_f(E * 48);
    float* vB     = alloc_f(E * 48);
    float* evt    = alloc_f(E * 6);
    int*   rev    = alloc_i(E);
    int*   hkeys  = alloc_i(HASH_SIZE);
    int*   hvals  = alloc_i(HASH_SIZE);

    // ---- launch helpers
    auto blks = [](long long n, int b) { return dim3((unsigned)((n + b - 1) / b)); };
    auto gemm = [&](GSeg g0, GSeg g1, GSeg g2, GSeg g3, int nseg,
                    const Pgvp& p, float* C, long long M, int N, int ldc, int relu) {
        GSeg gs[4] = {g0, g1, g2, g3};
        ASpec a{};
        int Kp = 0;
        for (int s = 0; s < nseg; ++s) {
            int wp = (gs[s].width + 3) & ~3;
            a.seg[s].base = gs[s].base; a.seg[s].gather = gs[s].gather;
            a.seg[s].stride = gs[s].stride; a.seg[s].wpad = wp;
            Kp += wp;
        }
        int Npad = (N + 31) & ~31;
        float* Wp2 = alloc_f((size_t)Kp * Npad);
        k_pack_w<<<blks((long long)Kp * Npad, 256), 256, 0, stream>>>(
            p.ws_w, N, Npad,
            gs[0].width, nseg > 1 ? gs[1].width : 0,
            nseg > 2 ? gs[2].width : 0, nseg > 3 ? gs[3].width : 0, Wp2, Kp);
        dim3 grid((unsigned)(Npad / 32), (unsigned)((M + 63) / 64));
        if (nseg == 2)
            k_gemm_wmma<2, 2><<<grid, 128, 0, stream>>>(a, Wp2, p.ws_b, C, M, N, Npad, ldc, relu);
        else
            k_gemm_wmma<4, 2><<<grid, 128, 0, stream>>>(a, Wp2, p.ws_b, C, M, N, Npad, ldc, relu);
    };
    GSeg gnul{nullptr, nullptr, 0, 0};
    auto gsg = [](const float* b, const long long* g, int wd, int st) {
        GSeg s; s.base = b; s.gather = g; s.width = wd; s.stride = st; return s;
    };
    VSeg vnul{nullptr, nullptr, 0};
    auto vsg = [](const float* b, const long long* g, int wd) {
        VSeg s; s.base = b; s.gather = g; s.width = wd; return s;
    };
    auto vecmm = [&](VSeg g0, VSeg g1, VSeg g2, int VI, const float* Wm, int VO,
                     float* outp, long long M, int act) {
        k_vec_mm<<<blks(M * VO, 256), 256, 0, stream>>>(g0, g1, g2, VI, Wm, VO, outp, M, act);
    };
    auto norms = [&](const float* vhp, float* vnp, long long c) {
        k_norm3<<<blks(c, 256), 256, 0, stream>>>(vhp, vnp, c);
    };
    auto lns = [&](float* s, const Pln& l, long long M, int S) {
        k_ln_s<<<blks(M, 4), 128, 0, stream>>>(s, l.gamma, l.beta, M, S);
    };
    auto lnv = [&](float* v, long long M, int NV) {
        k_ln_v<<<blks(M, 256), 256, 0, stream>>>(v, M, NV);
    };
    auto fill = [&](float* p, long long n) { k_fill<<<blks(n, 256), 256, 0, stream>>>(p, n, 0.f); };
    auto addip = [&](float* d, const float* s, long long n) {
        k_add<<<blks(n, 256), 256, 0, stream>>>(d, s, n);
    };

    // ================= input embeddings =================
    hipMemcpyAsync(hVsp, hVs, (size_t)(NN * 6) * sizeof(float),
                   hipMemcpyDeviceToDevice, stream);
    // W_v (nodes)
    vecmm(vsg(hVv, nullptr, 3), vnul, vnul, 3, W_v.wh, 16, nvh, NN, 0);
    norms(nvh, nvn, NN * 16);
    gemm(gsg(hVsp, nullptr, 6, 6), gsg(nvn, nullptr, 16, 16), gnul, gnul, 2,
         W_v, s_node, NN, 100, 100, 0);
    vecmm(vsg(nvh, nullptr, 16), vnul, vnul, 16, W_v.wv, 16, v_node, NN, 0);
    lns(s_node, ln_v, NN, 100);
    lnv(v_node, NN, 16);
    // W_e (edges); hEs width 32 is 4-aligned -> no overread of external input
    vecmm(vsg(hEv, nullptr, 1), vnul, vnul, 1, W_e.wh, 1, vh, E, 0);
    norms(vh, vn, E);
    gemm(gsg(hEs, nullptr, 32, 32), gsg(vn, nullptr, 1, 1), gnul, gnul, 2,
         W_e, es, E, 32, 32, 0);
    vecmm(vsg(vh, nullptr, 1), vnul, vnul, 1, W_e.wv, 1, ev, E, 0);
    lns(es, ln_e, E, 32);
    lnv(ev, E, 1);

    // reverse-edge index
    k_hash_init<<<blks(HASH_SIZE, 256), 256, 0, stream>>>(hkeys, HASH_SIZE);
    k_hash_insert<<<blks(E, 256), 256, 0, stream>>>(src, dst, E, NN, hkeys, hvals);
    k_hash_lookup<<<blks(E, 256), 256, 0, stream>>>(src, dst, E, NN, hkeys, hvals, rev);

    // ================= message-passing layers =================
    for (int L = 0; L < 2; ++L) {
        const NL& n_ = nl[L];
        const EL& e_ = el[L];
        // ---------- node conv ----------
        vecmm(vsg(v_node, src, 16), vsg(ev, nullptr, 1), vsg(v_node, dst, 16),
              33, n_.m0.wh, 33, vh, E, 0);
        norms(vh, vn, E * 33);
        gemm(gsg(s_node, src, 100, 100), gsg(es, nullptr, 32, 32),
             gsg(s_node, dst, 100, 100), gsg(vn, nullptr, 33, 33), 4,
             n_.m0, sA, E, 100, 100, 1);
        vecmm(vsg(vh, nullptr, 33), vnul, vnul, 33, n_.m0.wv, 16, vA, E, 1);

        vecmm(vsg(vA, nullptr, 16), vnul, vnul, 16, n_.m1.wh, 16, vh, E, 0);
        norms(vh, vn, E * 16);
        gemm(gsg(sA, nullptr, 100, 100), gsg(vn, nullptr, 16, 16), gnul, gnul, 2,
             n_.m1, sB, E, 100, 100, 1);
        vecmm(vsg(vh, nullptr, 16), vnul, vnul, 16, n_.m1.wv, 16, vB, E, 1);

        vecmm(vsg(vB, nullptr, 16), vnul, vnul, 16, n_.m2.wh, 16, vh, E, 0);
        norms(vh, vn, E * 16);
        gemm(gsg(sB, nullptr, 100, 100), gsg(vn, nullptr, 16, 16), gnul, gnul, 2,
             n_.m2, sA, E, 100, 100, 0);
        vecmm(vsg(vh, nullptr, 16), vnul, vnul, 16, n_.m2.wv, 16, vA, E, 0);

        fill(agg_s, NN * 100); fill(agg_v, NN * 48); fill(deg, NN);
        k_deg<<<blks(E, 256), 256, 0, stream>>>(dst, deg, E);
        k_scatter<<<blks(E * 100, 256), 256, 0, stream>>>(sA, dst, agg_s, E, 100);
        k_scatter<<<blks(E * 48, 256), 256, 0, stream>>>(vA, dst, agg_v, E, 48);
        k_resid<<<blks(NN * 100, 256), 256, 0, stream>>>(s_node, agg_s, deg, NN, 100);
        k_resid<<<blks(NN * 48, 256), 256, 0, stream>>>(v_node, agg_v, deg, NN, 48);
        lns(s_node, n_.ln0, NN, 100);
        lnv(v_node, NN, 16);

        // node feed-forward
        vecmm(vsg(v_node, nullptr, 16), vnul, vnul, 16, n_.ff0.wh, 32, nvh, NN, 0);
        norms(nvh, nvn, NN * 32);
        gemm(gsg(s_node, nullptr, 100, 100), gsg(nvn, nullptr, 32, 32), gnul, gnul, 2,
             n_.ff0, nfs, NN, 400, 400, 1);
        vecmm(vsg(nvh, nullptr, 32), vnul, vnul, 32, n_.ff0.wv, 32, nfv, NN, 1);
        vecmm(vsg(nfv, nullptr, 32), vnul, vnul, 32, n_.ff1.wh, 32, nvh, NN, 0);
        norms(nvh, nvn, NN * 32);
        gemm(gsg(nfs, nullptr, 400, 400), gsg(nvn, nullptr, 32, 32), gnul, gnul, 2,
             n_.ff1, nfs2, NN, 100, 100, 0);
        vecmm(vsg(nvh, nullptr, 32), vnul, vnul, 32, n_.ff1.wv, 16, nfv2, NN, 0);
        addip(s_node, nfs2, NN * 100);
        addip(v_node, nfv2, NN * 48);
        lns(s_node, n_.ln1, NN, 100);
        lnv(v_node, NN, 16);

        // ---------- edge layer ----------
        vecmm(vsg(v_node, src, 16), vsg(ev, nullptr, 1), vsg(v_node, dst, 16),
              33, e_.m0.wh, 33, vh, E, 0);
        norms(vh, vn, E * 33);
        gemm(gsg(s_node, src, 100, 100), gsg(es, nullptr, 32, 32),
             gsg(s_node, dst, 100, 100), gsg(vn, nullptr, 33, 33), 4,
             e_.m0, sA, E, 32, 32, 1);
        vecmm(vsg(vh, nullptr, 33), vnul, vnul, 33, e_.m0.wv, 1, vA, E, 1);

        vecmm(vsg(vA, nullptr, 1), vnul, vnul, 1, e_.m1.wh, 1, vh, E, 0);
        norms(vh, vn, E);
        gemm(gsg(sA, nullptr, 32, 32), gsg(vn, nullptr, 1, 1), gnul, gnul, 2,
             e_.m1, sB, E, 32, 32, 1);
        vecmm(vsg(vh, nullptr, 1), vnul, vnul, 1, e_.m1.wv, 1, vB, E, 1);

        vecmm(vsg(vB, nullptr, 1), vnul, vnul, 1, e_.m2.wh, 1, vh, E, 0);
        norms(vh, vn, E);
        gemm(gsg(sB, nullptr, 32, 32), gsg(vn, nullptr, 1, 1), gnul, gnul, 2,
             e_.m2, sA, E, 32, 32, 0);
        vecmm(vsg(vh, nullptr, 1), vnul, vnul, 1, e_.m2.wv, 1, vA, E, 0);

        // ds = 0.5*(ds + ds[rev])  (scalar channel only)
        hipMemcpyAsync(sB, sA, (size_t)(E * 32) * sizeof(float),
                       hipMemcpyDeviceToDevice, stream);
        k_merge_rev<<<blks(E * 32, 256), 256, 0, stream>>>(sA, sB, rev, E, 32);

        addip(es, sA, E * 32);
        addip(ev, vA, E * 3);
        lns(es, e_.ln0, E, 32);
        lnv(ev, E, 1);

        // edge feed-forward
        vecmm(vsg(ev, nullptr, 1), vnul, vnul, 1, e_.ff0.wh, 2, vh, E, 0);
        norms(vh, vn, E * 2);
        gemm(gsg(es, nullptr, 32, 32), gsg(vn, nullptr, 2, 2), gnul, gnul, 2,
             e_.ff0, sA, E, 128, 128, 1);
        vecmm(vsg(vh, nullptr, 2), vnul, vnul, 2, e_.ff0.wv, 2, evt, E, 1);
        vecmm(vsg(evt, nullptr, 2), vnul, vnul, 2, e_.ff1.wh, 2, vh, E, 0);
        norms(vh, vn, E * 2);
        gemm(gsg(sA, nullptr, 128, 128), gsg(vn, nullptr, 2, 2), gnul, gnul, 2,
             e_.ff1, sB, E, 32, 32, 0);
        vecmm(vsg(vh, nullptr, 2), vnul, vnul, 2, e_.ff1.wv, 1, vA, E, 0);
        addip(es, sB, E * 32);
        addip(ev, vA, E * 3);
        lns(es, e_.ln1, E, 32);
        lnv(ev, E, 1);
    }

    // ================= output head =================
    float* out = (float*)d_out;
    vecmm(vsg(ev, nullptr, 1), vnul, vnul, 1, W_out.wh, 1, vh, E, 0);
    norms(vh, vn, E);
    gemm(gsg(es, nullptr, 32, 32), gsg(vn, nullptr, 1, 1), gnul, gnul, 2,
         W_out, out, E, 400, 400, 0);
    k_merge_out<<<blks(E * 400, 256), 256, 0, stream>>>(out, rev, src, dst, E);
}